// test_22960895164753
// MI455X (gfx1250) — compile-verified
//
#include <hip/hip_runtime.h>

// LSTM (H=5, I=3, O=2) over S=2048 steps, B=4096 batch, fused FC head.
// One wave32 = 16 batch elements (lane pair n, n+16 owns batch n).
// Per step: z(20x16) = [W_hh | W_ih | b](20x9, K-padded to 12) x [h; x; 1]
// via 2 chains of 3 V_WMMA_F32_16X16X4_F32.
// Hidden-unit ownership: lo lane owns units {0,1,4}, hi lane owns {2,3},
// so next-step B1 = (hS0, hS1) on both halves with no cross-lane traffic
// on the recurrence critical path. Gate exchange uses v_permlanex16
// (ds_swizzle SWAPX16 fallback).

typedef float v2f __attribute__((ext_vector_type(2)));
typedef float v8f __attribute__((ext_vector_type(8)));

#define S_LEN 2048
#define B_DIM 4096
#define I_DIM 3
#define H_DIM 5
#define G_DIM 20  // 4*H
#define O_DIM 2

#define LOG2E 1.442695040888963f

__device__ __forceinline__ float swapx16(float x) {
#if __has_builtin(__builtin_amdgcn_permlanex16)
  // identity selectors -> lane n <-> lane n^16 (VALU crosslane, no LDS trip)
  return __int_as_float(__builtin_amdgcn_permlanex16(
      __float_as_int(x), __float_as_int(x), 0x76543210, 0xfedcba98, false, false));
#else
  // group-of-32 swizzle: and=0x1f, or=0, xor=0x10
  return __int_as_float(__builtin_amdgcn_ds_swizzle(__float_as_int(x), 0x401F));
#endif
}

__device__ __forceinline__ float fast_sigmoid(float x) {
  float e = __builtin_amdgcn_exp2f(x * -LOG2E);
  return __builtin_amdgcn_rcpf(1.0f + e);
}

__device__ __forceinline__ float fast_tanh(float x) {
  // tanh(x) = 2*sigmoid(2x) - 1
  float u = __builtin_amdgcn_rcpf(1.0f + __builtin_amdgcn_exp2f(x * (-2.0f * LOG2E)));
  return fmaf(2.0f, u, -1.0f);
}

__global__ __launch_bounds__(256) void lstm_wmma_kernel(
    const float* __restrict__ in,    // [S, B, I]
    const float* __restrict__ W_ih,  // [4H, I]
    const float* __restrict__ W_hh,  // [4H, H]
    const float* __restrict__ b_ih,  // [4H]
    const float* __restrict__ b_hh,  // [4H]
    const float* __restrict__ W_fc,  // [O, H]
    const float* __restrict__ b_fc,  // [O]
    float* __restrict__ out)         // [S, B, O]
{
  const int lane = threadIdx.x & 31;
  const int wave = threadIdx.x >> 5;
  const int tile = blockIdx.x * (blockDim.x >> 5) + wave;  // 16-batch tile
  const int row  = lane & 15;
  const bool lo  = lane < 16;
  const int bb   = tile * 16 + row;  // batch element owned by lane pair
  const int c0   = lo ? 0 : 2;       // A-matrix K columns held by this lane

  // Extended-K column value for gate g: cols 0..4 = W_hh, 5..7 = W_ih,
  // 8 = (b_ih + b_hh), 9..11 = zero padding.
  auto colval = [&](int g, int c) -> float {
    if (g >= G_DIM) return 0.0f;
    if (c < 5) return W_hh[g * H_DIM + c];
    if (c < 8) return W_ih[g * I_DIM + (c - 5)];
    if (c == 8) return b_ih[g] + b_hh[g];
    return 0.0f;
  };

  // A matrices (wave-invariant), 16x4 f32 layout:
  // lane m: vgpr0=A[m][K], vgpr1=A[m][K+1], K = 0 (lanes 0-15) / 2 (lanes 16-31)
  const int g1 = row;       // M-tile over gates 0..15
  const int g2 = 16 + row;  // M-tile over gates 16..19 (rows 4..15 zero)
  v2f A1 = { colval(g1, c0),     colval(g1, c0 + 1) };   // K 0..3
  v2f A2 = { colval(g1, c0 + 4), colval(g1, c0 + 5) };   // K 4..7
  v2f A3 = { colval(g1, c0 + 8), colval(g1, c0 + 9) };   // K 8..11
  v2f A4 = { colval(g2, c0),     colval(g2, c0 + 1) };
  v2f A5 = { colval(g2, c0 + 4), colval(g2, c0 + 5) };
  v2f A6 = { colval(g2, c0 + 8), colval(g2, c0 + 9) };

  // FC head with per-half preselected weights.
  // Slot ownership: lo = units {0,1,4}, hi = units {2,3}.
  // u_k = ws0*hS0 + ws1*hS1 + ws2*hS2 + (lo ? b_k : 0); y_k = sigmoid(u_k + swap(u_k))
  const float ws00 = lo ? W_fc[0 * H_DIM + 0] : W_fc[0 * H_DIM + 2];
  const float ws01 = lo ? W_fc[0 * H_DIM + 1] : W_fc[0 * H_DIM + 3];
  const float ws02 = lo ? W_fc[0 * H_DIM + 4] : 0.0f;
  const float ws10 = lo ? W_fc[1 * H_DIM + 0] : W_fc[1 * H_DIM + 2];
  const float ws11 = lo ? W_fc[1 * H_DIM + 1] : W_fc[1 * H_DIM + 3];
  const float ws12 = lo ? W_fc[1 * H_DIM + 4] : 0.0f;
  const float bs0  = lo ? b_fc[0] : 0.0f;
  const float bs1  = lo ? b_fc[1] : 0.0f;

  // Branchless mixed activation for D12 vgprs 2..6:
  // lo holds gates 2..6 (sigmoid), hi holds gates 10..14 (tanh).
  const float kAct = lo ? -LOG2E : -2.0f * LOG2E;
  const float aAct = lo ? 1.0f : 2.0f;
  const float cAct = lo ? 0.0f : -1.0f;

  // Per-lane recurrent state: lo = (h0,c0),(h1,c1),(h4,c4); hi = (h2,c2),(h3,c3),-
  float hS0 = 0.f, hS1 = 0.f, hS2 = 0.f;
  float cS0 = 0.f, cS1 = 0.f, cS2 = 0.f;

  const long xstep = (long)B_DIM * I_DIM;
  const long ostep = (long)B_DIM * O_DIM;
  const long xbase = (long)bb * I_DIM;
  const long obase = (long)bb * O_DIM;

  for (int t = 0; t < S_LEN; ++t) {
    const float* xp = in + (long)t * xstep + xbase;
    float x0 = xp[0], x1 = xp[1], x2 = xp[2];
    if (t + 8 < S_LEN) __builtin_prefetch(xp + 8 * xstep, 0, 1);

    // B matrix (K x 16): lo lanes hold K,K+1 = {0,1},{4,5},{8,9}; hi = {2,3},{6,7},{10,11}
    v2f B1, B2, B3;
    B1.x = hS0;              B1.y = hS1;           // K0,K1 = h0,h1 ; K2,K3 = h2,h3
    B2.x = lo ? hS2 : x1;    B2.y = lo ? x0 : x2;  // K4,K5 = h4,x0 ; K6,K7 = x1,x2
    B3.x = lo ? 1.0f : 0.0f; B3.y = 0.0f;          // K8 = 1 (bias), K9..11 = 0

    v8f z12 = {};
    z12 = __builtin_amdgcn_wmma_f32_16x16x4_f32(false, A1, false, B1, (short)0, z12, false, false);
    z12 = __builtin_amdgcn_wmma_f32_16x16x4_f32(false, A2, false, B2, (short)0, z12, false, false);
    z12 = __builtin_amdgcn_wmma_f32_16x16x4_f32(false, A3, false, B3, (short)0, z12, false, false);
    v8f z34 = {};
    z34 = __builtin_amdgcn_wmma_f32_16x16x4_f32(false, A4, false, B1, (short)0, z34, false, false);
    z34 = __builtin_amdgcn_wmma_f32_16x16x4_f32(false, A5, false, B2, (short)0, z34, false, false);
    z34 = __builtin_amdgcn_wmma_f32_16x16x4_f32(false, A6, false, B3, (short)0, z34, false, false);

    // D layout: lo lane vgpr r = z[gate r]; hi lane vgpr r = z[gate 8+r].
    // Activate in place. lo: gates 0-7 all sigmoid. hi: f3,f4 (r=0,1) sigmoid,
    // g0..g4 (r=2..6) tanh, o0 (r=7) sigmoid.
    float a12[8];
    a12[0] = fast_sigmoid(z12[0]);
    a12[1] = fast_sigmoid(z12[1]);
#pragma unroll
    for (int r = 2; r <= 6; ++r) {
      float u = __builtin_amdgcn_rcpf(1.0f + __builtin_amdgcn_exp2f(z12[r] * kAct));
      a12[r] = fmaf(aAct, u, cAct);
    }
    a12[7] = fast_sigmoid(z12[7]);
    float ao1 = fast_sigmoid(z34[0]);  // gate16 = o1 (valid on lo)
    float ao2 = fast_sigmoid(z34[1]);  // gate17 = o2
    float ao3 = fast_sigmoid(z34[2]);  // gate18 = o3
    float ao4 = fast_sigmoid(z34[3]);  // gate19 = o4

    // Cross-lane exchange (each swap serves both halves):
    float sw1 = swapx16(a12[1]);  // lo<-f4(g9)      hi<-i1 (unused)
    float sw2 = swapx16(a12[2]);  // lo<-g0(g10)     hi<-i2(g2)
    float sw3 = swapx16(a12[3]);  // lo<-g1(g11)     hi<-i3(g3)
    float sw6 = swapx16(a12[6]);  // lo<-g4(g14)     hi<-f1 (unused)
    float sw7 = swapx16(a12[7]);  // lo<-o0(g15)     hi<-f2(g7)
    float swo2 = swapx16(ao2);    // hi<-o2(g17)     lo<-garbage
    float swo3 = swapx16(ao3);    // hi<-o3(g18)     lo<-garbage

    // Slot updates. slot0: lo=unit0, hi=unit2. slot1: lo=unit1, hi=unit3.
    // slot2: lo=unit4, hi=inactive (zeroed).
    float i0 = lo ? a12[0] : sw2;
    float f0 = lo ? a12[5] : sw7;
    float g0 = lo ? sw2    : a12[4];
    float o0 = lo ? sw7    : swo2;

    float i1 = lo ? a12[1] : sw3;
    float f1 = lo ? a12[6] : a12[0];
    float g1 = lo ? sw3    : a12[5];
    float o1 = lo ? ao1    : swo3;

    float i2 = lo ? a12[4] : 0.0f;
    float f2 = lo ? sw1    : 0.0f;
    float g2 = lo ? sw6    : 0.0f;
    float o2 = lo ? ao4    : 0.0f;

    cS0 = fmaf(f0, cS0, i0 * g0);
    cS1 = fmaf(f1, cS1, i1 * g1);
    cS2 = fmaf(f2, cS2, i2 * g2);
    hS0 = o0 * fast_tanh(cS0);
    hS1 = o1 * fast_tanh(cS1);
    hS2 = o2 * fast_tanh(cS2);

    // FC head: per-half partial sums, one swap-reduce, store on lo lanes.
    float u0 = fmaf(ws00, hS0, fmaf(ws01, hS1, fmaf(ws02, hS2, bs0)));
    float u1 = fmaf(ws10, hS0, fmaf(ws11, hS1, fmaf(ws12, hS2, bs1)));
    float y0 = fast_sigmoid(u0 + swapx16(u0));
    float y1 = fast_sigmoid(u1 + swapx16(u1));
    if (lo) {
      v2f y;
      y.x = y0;
      y.y = y1;
      __builtin_nontemporal_store(y, (v2f*)(out + (long)t * ostep + obase));
    }
  }
}

extern "C" void kernel_launch(void* const* d_in, const int* in_sizes, int n_in,
                              void* d_out, int out_size, void* d_ws, size_t ws_size,
                              hipStream_t stream) {
  (void)in_sizes; (void)n_in; (void)out_size; (void)d_ws; (void)ws_size;
  const float* in   = (const float*)d_in[0];
  const float* W_ih = (const float*)d_in[1];
  const float* W_hh = (const float*)d_in[2];
  const float* b_ih = (const float*)d_in[3];
  const float* b_hh = (const float*)d_in[4];
  const float* W_fc = (const float*)d_in[5];
  const float* b_fc = (const float*)d_in[6];
  float* out = (float*)d_out;

  const int tiles = B_DIM / 16;               // 256 waves
  const int threads = 256;                    // 8 waves / block
  const int blocks = tiles / (threads / 32);  // 32 blocks
  lstm_wmma_kernel<<<blocks, threads, 0, stream>>>(in, W_ih, W_hh, b_ih, b_hh,
                                                   W_fc, b_fc, out);
}